// PivotGCN_6356551598511
// MI455X (gfx1250) — compile-verified
//
#include <hip/hip_runtime.h>
#include <math.h>

typedef __attribute__((ext_vector_type(2))) float v2f;
typedef __attribute__((ext_vector_type(8))) float v8f;

#define DIM   64   // input dim D
#define HID   64   // hidden dim H
#define ODIM  8    // output dim

// ---------------------------------------------------------------------------
// Hardware f32 atomic add: non-returning GLOBAL_ATOMIC_ADD_F32 at device
// scope (RMW resolves at L2 atomic units, coherent across WGPs). Tracked on
// STOREcnt -> fire-and-forget, no CAS loop, no data return.
// ---------------------------------------------------------------------------
__device__ __forceinline__ void hw_atomic_add_f32(float* p, float v) {
    asm volatile("global_atomic_add_f32 %0, %1, off scope:SCOPE_DEV"
                 :: "v"(p), "v"(v)
                 : "memory");
}

// ---------------------------------------------------------------------------
// 1) deg[i] = 1.0 (self loop); zero pooled accumulators
// ---------------------------------------------------------------------------
__global__ void pg_init_kernel(float* __restrict__ deg, float* __restrict__ gemb,
                               float* __restrict__ counts, int n, int g) {
    int i = blockIdx.x * blockDim.x + threadIdx.x;
    if (i < n)        deg[i]    = 1.0f;
    if (i < g * HID)  gemb[i]   = 0.0f;
    if (i < g)        counts[i] = 0.0f;
}

// ---------------------------------------------------------------------------
// 2) degree over source rows
// ---------------------------------------------------------------------------
__global__ void pg_deg_kernel(const int* __restrict__ ei, float* __restrict__ deg, int e) {
    int i = blockIdx.x * blockDim.x + threadIdx.x;
    if (i < e) hw_atomic_add_f32(&deg[ei[i]], 1.0f);
}

// ---------------------------------------------------------------------------
// 3) dis = deg^-0.5 (in place)
// ---------------------------------------------------------------------------
__global__ void pg_rsqrt_kernel(float* __restrict__ deg, int n) {
    int i = blockIdx.x * blockDim.x + threadIdx.x;
    if (i < n) deg[i] = rsqrtf(deg[i]);
}

// ---------------------------------------------------------------------------
// 4) self-loop contribution (full write initializes aggr, no atomics needed)
//    aggr[i,d] = dis[i]^2 * x[i,d]
// ---------------------------------------------------------------------------
__global__ void pg_selfloop_kernel(const float* __restrict__ x, const float* __restrict__ dis,
                                   float* __restrict__ aggr, int n) {
    int idx = blockIdx.x * blockDim.x + threadIdx.x;
    if (idx < n * DIM) {
        int i   = idx >> 6;
        float d = dis[i];
        aggr[idx] = d * d * x[idx];
    }
}

// ---------------------------------------------------------------------------
// 5) edge scatter: one wave32 per edge, 2 dims per lane
//    aggr[col, :] += ea * dis[row]*dis[col] * x[row, :]
// ---------------------------------------------------------------------------
__global__ void pg_scatter_kernel(const float* __restrict__ x, const int* __restrict__ ei,
                                  const float* __restrict__ ea, const float* __restrict__ dis,
                                  float* __restrict__ aggr, int e) {
    int wave = blockIdx.x * (blockDim.x >> 5) + (threadIdx.x >> 5);
    int lane = threadIdx.x & 31;
    if (wave >= e) return;               // wave-uniform exit
    int r = ei[wave];
    int c = ei[e + wave];
    float coeff = ea[wave] * dis[r] * dis[c];
    const float* xr = x + (size_t)r * DIM;
    float* ac = aggr + (size_t)c * DIM;
    hw_atomic_add_f32(&ac[lane],      coeff * xr[lane]);
    hw_atomic_add_f32(&ac[lane + 32], coeff * xr[lane + 32]);
}

// ---------------------------------------------------------------------------
// 6) h = tanh(aggr @ W1^T + b1) via V_WMMA_F32_16X16X4_F32
//    One wave computes one 16x16 tile of h. K=64 -> 16 WMMA issues.
//    A (16x4 f32) layout: lanes 0-15 carry K=0..1, lanes 16-31 K=2..3
//    (ISA 7.12.2); B 4x16 mirrors that. C/D: VGPR v, lanes 0-15 -> M=v,
//    lanes 16-31 -> M=v+8.
// ---------------------------------------------------------------------------
__global__ void pg_gemm1_tanh_kernel(const float* __restrict__ A,   // aggr [N,64]
                                     const float* __restrict__ W,   // W1 [64,64]
                                     const float* __restrict__ bias,// [64]
                                     float* __restrict__ Hout,      // [N,64]
                                     int nRowTiles, int n) {
    int wave = blockIdx.x * (blockDim.x >> 5) + (threadIdx.x >> 5);
    int nWaves = nRowTiles * (HID / 16);
    if (wave >= nWaves) return;          // wave-uniform exit
    int colTile = wave & 3;              // HID/16 == 4
    int rowTile = wave >> 2;
    int m0 = rowTile * 16, n0 = colTile * 16;
    int lane  = threadIdx.x & 31;
    int idx16 = lane & 15;
    int half  = lane >> 4;
    int koff  = half << 1;

    int mload = m0 + idx16; if (mload > n - 1) mload = n - 1;   // clamp (lane-local select)
    const float* arow = A + (size_t)mload * DIM;
    const float* brow = W + (size_t)(n0 + idx16) * DIM;

    v8f c = {};
#pragma unroll
    for (int k0 = 0; k0 < DIM; k0 += 4) {
        float2 af = *reinterpret_cast<const float2*>(arow + k0 + koff);
        float2 bf = *reinterpret_cast<const float2*>(brow + k0 + koff);
        v2f a; a[0] = af.x; a[1] = af.y;
        v2f b; b[0] = bf.x; b[1] = bf.y;
        c = __builtin_amdgcn_wmma_f32_16x16x4_f32(false, a, false, b, (short)0, c,
                                                  false, false);
    }
    float bval = bias[n0 + idx16];
    int mbase = m0 + (half << 3);
#pragma unroll
    for (int v = 0; v < 8; ++v) {
        int m = mbase + v;
        if (m < n) Hout[(size_t)m * HID + n0 + idx16] = tanhf(c[v] + bval);
    }
}

// ---------------------------------------------------------------------------
// 7) out = sigmoid(h @ W2^T + b2)  (OUT=8; pad B cols 8..15 with zeros)
// ---------------------------------------------------------------------------
__global__ void pg_gemm2_sig_kernel(const float* __restrict__ Hin,  // [N,64]
                                    const float* __restrict__ W2,   // [8,64]
                                    const float* __restrict__ b2,   // [8]
                                    float* __restrict__ Out,        // [N,8]
                                    int nRowTiles, int n) {
    int wave = blockIdx.x * (blockDim.x >> 5) + (threadIdx.x >> 5);
    if (wave >= nRowTiles) return;       // wave-uniform exit
    int m0 = wave * 16;
    int lane  = threadIdx.x & 31;
    int idx16 = lane & 15;
    int half  = lane >> 4;
    int koff  = half << 1;

    int mload = m0 + idx16; if (mload > n - 1) mload = n - 1;
    const float* arow = Hin + (size_t)mload * HID;
    bool nvalid = idx16 < ODIM;
    const float* brow = W2 + (size_t)(nvalid ? idx16 : 0) * HID;

    v8f c = {};
#pragma unroll
    for (int k0 = 0; k0 < HID; k0 += 4) {
        float2 af = *reinterpret_cast<const float2*>(arow + k0 + koff);
        float2 bf = *reinterpret_cast<const float2*>(brow + k0 + koff);
        v2f a; a[0] = af.x; a[1] = af.y;
        v2f b; b[0] = nvalid ? bf.x : 0.0f; b[1] = nvalid ? bf.y : 0.0f;
        c = __builtin_amdgcn_wmma_f32_16x16x4_f32(false, a, false, b, (short)0, c,
                                                  false, false);
    }
    if (nvalid) {
        float bval = b2[idx16];
        int mbase = m0 + (half << 3);
#pragma unroll
        for (int v = 0; v < 8; ++v) {
            int m = mbase + v;
            if (m < n) {
                float s = c[v] + bval;
                Out[(size_t)m * ODIM + idx16] = 1.0f / (1.0f + expf(-s));
            }
        }
    }
}

// ---------------------------------------------------------------------------
// 8) mean-pool accumulate: one wave per node, 2 dims per lane
// ---------------------------------------------------------------------------
__global__ void pg_pool_kernel(const float* __restrict__ Hin, const int* __restrict__ batch,
                               float* __restrict__ gemb, float* __restrict__ counts, int n) {
    int wave = blockIdx.x * (blockDim.x >> 5) + (threadIdx.x >> 5);
    int lane = threadIdx.x & 31;
    if (wave >= n) return;
    int g = batch[wave];
    float2 hv = *reinterpret_cast<const float2*>(&Hin[(size_t)wave * HID + lane * 2]);
    hw_atomic_add_f32(&gemb[(size_t)g * HID + lane * 2],     hv.x);
    hw_atomic_add_f32(&gemb[(size_t)g * HID + lane * 2 + 1], hv.y);
    if (lane == 0) hw_atomic_add_f32(&counts[g], 1.0f);
}

// ---------------------------------------------------------------------------
// 9) optimal = sigmoid((gemb/count) @ Wopt^T + bopt)   [G,1]
// ---------------------------------------------------------------------------
__global__ void pg_optimal_kernel(const float* __restrict__ gemb, const float* __restrict__ counts,
                                  const float* __restrict__ Wopt, const float* __restrict__ bopt,
                                  float* __restrict__ OutOpt, int g) {
    int i = blockIdx.x * blockDim.x + threadIdx.x;
    if (i >= g) return;
    float inv = 1.0f / fmaxf(counts[i], 1.0f);
    float acc = 0.0f;
#pragma unroll 8
    for (int d = 0; d < HID; ++d) acc += gemb[(size_t)i * HID + d] * Wopt[d];
    float s = acc * inv + bopt[0];
    OutOpt[i] = 1.0f / (1.0f + expf(-s));
}

// ---------------------------------------------------------------------------
extern "C" void kernel_launch(void* const* d_in, const int* in_sizes, int n_in,
                              void* d_out, int out_size, void* d_ws, size_t ws_size,
                              hipStream_t stream) {
    const float* x    = (const float*)d_in[0];   // [N,64]
    const int*   ei   = (const int*)  d_in[1];   // [2,E]
    const float* ea   = (const float*)d_in[2];   // [E]
    const int*   bat  = (const int*)  d_in[3];   // [N]
    const float* W1   = (const float*)d_in[4];   // [64,64]
    const float* b1   = (const float*)d_in[5];   // [64]
    const float* W2   = (const float*)d_in[6];   // [8,64]
    const float* b2   = (const float*)d_in[7];   // [8]
    const float* Wopt = (const float*)d_in[8];   // [1,64]
    const float* bopt = (const float*)d_in[9];   // [1]

    const int N = in_sizes[0] / DIM;
    const int E = in_sizes[2];
    const int G = out_size - N * ODIM;

    // workspace layout (floats), deg region padded to 256 floats
    float* ws = (float*)d_ws;
    size_t off = 0;
    float* deg    = ws + off; off += ((size_t)N + 255) & ~(size_t)255;      // deg -> dis
    float* aggr   = ws + off; off += (size_t)N * DIM;
    float* hbuf   = ws + off; off += (size_t)N * HID;
    float* gemb   = ws + off; off += (size_t)G * HID;
    float* counts = ws + off; off += (size_t)G;
    (void)ws_size; (void)n_in;

    float* outNode = (float*)d_out;              // [N,8]
    float* outOpt  = outNode + (size_t)N * ODIM; // [G]

    const int B = 256;                           // 8 waves per block

    pg_init_kernel<<<(N + B - 1) / B, B, 0, stream>>>(deg, gemb, counts, N, G);
    pg_deg_kernel<<<(E + B - 1) / B, B, 0, stream>>>(ei, deg, E);
    pg_rsqrt_kernel<<<(N + B - 1) / B, B, 0, stream>>>(deg, N);
    pg_selfloop_kernel<<<((size_t)N * DIM + B - 1) / B, B, 0, stream>>>(x, deg, aggr, N);

    // one wave per edge
    pg_scatter_kernel<<<(E + 7) / 8, B, 0, stream>>>(x, ei, ea, deg, aggr, E);

    // GEMM1 + tanh : (N/16)*(H/16) waves
    int rowTiles = (N + 15) / 16;
    int waves1 = rowTiles * (HID / 16);
    pg_gemm1_tanh_kernel<<<(waves1 + 7) / 8, B, 0, stream>>>(aggr, W1, b1, hbuf, rowTiles, N);

    // GEMM2 + sigmoid : one wave per 16-row tile
    pg_gemm2_sig_kernel<<<(rowTiles + 7) / 8, B, 0, stream>>>(hbuf, W2, b2, outNode, rowTiles, N);

    // mean pool accumulate + final head
    pg_pool_kernel<<<(N + 7) / 8, B, 0, stream>>>(hbuf, bat, gemb, counts, N);
    pg_optimal_kernel<<<(G + 127) / 128, 128, 0, stream>>>(gemb, counts, Wopt, bopt, outOpt, G);
}